// SSIMLoss_39333310497153
// MI455X (gfx1250) — compile-verified
//
#include <hip/hip_runtime.h>
#include <stdint.h>

// ---------------- problem geometry ----------------
#define IMG_H 512
#define IMG_W 512
#define NIMG  48          // B*C = 16*3
#define OUT_H 510
#define OUT_W 510
#define TILE_H 16         // output rows per block
#define TILE_W 64         // output cols per block
#define IT_H   (TILE_H + 2)   // 18 input rows
#define IT_W   (TILE_W + 2)   // 66 input cols (used by compute stages)
#define IT_WP  68             // padded LDS row stride (17 x 16B chunks, 16B aligned)
#define CHUNKS_PER_ROW 17     // 68/4 b128 chunks per tile row
#define NCHUNK (IT_H * CHUNKS_PER_ROW)      // 306 chunks per tensor
#define TILES_X 8         // ceil(510/64)
#define TILES_Y 32        // ceil(510/16)
#define NPART  (TILES_X * TILES_Y * NIMG)   // 12288 partial sums
#define NVALID ((float)(NIMG) * (float)OUT_H * (float)OUT_W) // 12,484,800

typedef int v4i __attribute__((__vector_size__(4 * sizeof(int))));
typedef __attribute__((address_space(1))) v4i gas_v4i;
typedef __attribute__((address_space(3))) v4i las_v4i;

// ---- CDNA5 async global->LDS copy, 16B per lane (bypasses VGPRs, ASYNCcnt) ----
__device__ __forceinline__ void async_ld_b128(const float* gsrc, float* ldst) {
#if __has_builtin(__builtin_amdgcn_global_load_async_to_lds_b128)
  __builtin_amdgcn_global_load_async_to_lds_b128(
      (gas_v4i*)(uintptr_t)gsrc,
      (las_v4i*)(uint32_t)(uintptr_t)ldst,   // low 32 bits of generic LDS ptr = LDS offset
      0, 0);
#else
  uint32_t loff = (uint32_t)(uintptr_t)ldst;
  asm volatile("global_load_async_to_lds_b128 %0, %1, off"
               :: "v"(loff), "v"(gsrc) : "memory");
#endif
}

__device__ __forceinline__ void wait_async0() {
#if __has_builtin(__builtin_amdgcn_s_wait_asynccnt)
  __builtin_amdgcn_s_wait_asynccnt(0);
#else
  asm volatile("s_wait_asynccnt 0" ::: "memory");
#endif
}

// ---------------- main SSIM tile kernel ----------------
__global__ __launch_bounds__(256)
void ssim_tile_kernel(const float* __restrict__ img1,
                      const float* __restrict__ img2,
                      float* __restrict__ partial) {
  __shared__ alignas(16) float sx[IT_H * IT_WP];   // raw x tile (padded stride)
  __shared__ alignas(16) float sy[IT_H * IT_WP];   // raw y tile
  __shared__ float csx [TILE_H * IT_W];            // vertical 3-sums
  __shared__ float csy [TILE_H * IT_W];
  __shared__ float csxx[TILE_H * IT_W];
  __shared__ float csyy[TILE_H * IT_W];
  __shared__ float csxy[TILE_H * IT_W];
  __shared__ float red[256];

  const int tid = threadIdx.x;
  const int r0 = blockIdx.y * TILE_H;        // input-tile row origin
  const int c0 = blockIdx.x * TILE_W;        // input-tile col origin
  const size_t ibase = (size_t)blockIdx.z * (size_t)(IMG_H * IMG_W);

  // ---- Stage A: async b128 loads of 18x68 tiles of x and y into LDS ----
  // Each chunk = 4 consecutive floats, 16B aligned in global and LDS.
  // Row clamp (<=511) and chunk-base clamp (<=508) keep all reads in-bounds;
  // clamped duplicates land only in halo/pad columns used by masked outputs.
  for (int e = tid; e < NCHUNK; e += 256) {
    int lr = e / CHUNKS_PER_ROW;
    int ck = e - lr * CHUNKS_PER_ROW;
    int gr = r0 + lr;     if (gr > IMG_H - 1) gr = IMG_H - 1;
    int gc = c0 + 4 * ck; if (gc > IMG_W - 4) gc = IMG_W - 4;
    size_t off = ibase + (size_t)gr * IMG_W + (size_t)gc;
    int ldsoff = lr * IT_WP + 4 * ck;
    async_ld_b128(img1 + off, &sx[ldsoff]);
    async_ld_b128(img2 + off, &sy[ldsoff]);
  }
  wait_async0();          // this wave's async LDS writes complete
  __syncthreads();        // all waves' writes visible

  // ---- Stage B: vertical 3-row sums of x, y, x^2, y^2, xy ----
  for (int e = tid; e < TILE_H * IT_W; e += 256) {
    int orow = e / IT_W;
    int lc   = e - orow * IT_W;
    int b    = orow * IT_WP + lc;
    float x0 = sx[b], x1 = sx[b + IT_WP], x2 = sx[b + 2 * IT_WP];
    float y0 = sy[b], y1 = sy[b + IT_WP], y2 = sy[b + 2 * IT_WP];
    csx [e] = x0 + x1 + x2;
    csy [e] = y0 + y1 + y2;
    csxx[e] = fmaf(x2, x2, fmaf(x1, x1, x0 * x0));
    csyy[e] = fmaf(y2, y2, fmaf(y1, y1, y0 * y0));
    csxy[e] = fmaf(x2, y2, fmaf(x1, y1, x0 * y0));
  }
  __syncthreads();

  // ---- Stage C: horizontal 3-sums + SSIM map, 4 outputs/thread ----
  // Folded constants (WIN=3, NPIX=9, COV_NORM=9/8, C1=1e-4, C2=9e-4):
  //   A1 = (2/81)*Sx*Sy + C1
  //   A2 = Sxy/4 - Sx*Sy/36 + C2
  //   B1 = (Sx^2+Sy^2)/81 + C1
  //   B2 = (Sxx+Syy)/8 - (Sx^2+Sy^2)/72 + C2
  const float C1 = 1.0e-4f;
  const float C2 = 9.0e-4f;
  float acc = 0.0f;
  {
    const int row = tid >> 4;            // 0..15
    const int cb  = (tid & 15) << 2;     // 0,4,...,60
    const int rb  = row * IT_W + cb;
    float ax[6], ay[6], axx[6], ayy[6], axy[6];
#pragma unroll
    for (int j = 0; j < 6; ++j) {
      ax[j]  = csx [rb + j];
      ay[j]  = csy [rb + j];
      axx[j] = csxx[rb + j];
      ayy[j] = csyy[rb + j];
      axy[j] = csxy[rb + j];
    }
    const int org = r0 + row;            // global output row
    const int ocg = c0 + cb;             // global output col base
#pragma unroll
    for (int i = 0; i < 4; ++i) {
      float Sx  = ax [i] + ax [i + 1] + ax [i + 2];
      float Sy  = ay [i] + ay [i + 1] + ay [i + 2];
      float Sxx = axx[i] + axx[i + 1] + axx[i + 2];
      float Syy = ayy[i] + ayy[i + 1] + ayy[i + 2];
      float Sxy = axy[i] + axy[i + 1] + axy[i + 2];
      float SxSy = Sx * Sy;
      float P  = fmaf(Sx, Sx, Sy * Sy);
      float Q  = Sxx + Syy;
      float A1 = fmaf(SxSy, 2.0f / 81.0f, C1);
      float A2 = fmaf(Sxy, 0.25f, fmaf(SxSy, -1.0f / 36.0f, C2));
      float B1 = fmaf(P, 1.0f / 81.0f, C1);
      float B2 = fmaf(Q, 0.125f, fmaf(P, -1.0f / 72.0f, C2));
      float S  = (A1 * A2) * __builtin_amdgcn_rcpf(B1 * B2);
      bool valid = (org < OUT_H) && (ocg + i < OUT_W);
      acc += valid ? S : 0.0f;
    }
  }

  // ---- deterministic in-block tree reduction ----
  red[tid] = acc;
  __syncthreads();
#pragma unroll
  for (int s = 128; s > 0; s >>= 1) {
    if (tid < s) red[tid] += red[tid + s];
    __syncthreads();
  }
  if (tid == 0) {
    int pidx = (blockIdx.z * TILES_Y + blockIdx.y) * TILES_X + blockIdx.x;
    partial[pidx] = red[0];
  }
}

// ---------------- deterministic finalize: 1 - mean(S) ----------------
__global__ __launch_bounds__(256)
void ssim_finalize_kernel(const float* __restrict__ partial,
                          float* __restrict__ out) {
  __shared__ float red[256];
  float s = 0.0f;
  for (int i = threadIdx.x; i < NPART; i += 256) s += partial[i];
  red[threadIdx.x] = s;
  __syncthreads();
#pragma unroll
  for (int k = 128; k > 0; k >>= 1) {
    if (threadIdx.x < k) red[threadIdx.x] += red[threadIdx.x + k];
    __syncthreads();
  }
  if (threadIdx.x == 0) out[0] = 1.0f - red[0] * (1.0f / NVALID);
}

extern "C" void kernel_launch(void* const* d_in, const int* in_sizes, int n_in,
                              void* d_out, int out_size, void* d_ws, size_t ws_size,
                              hipStream_t stream) {
  (void)in_sizes; (void)n_in; (void)out_size; (void)ws_size;
  const float* img1 = (const float*)d_in[0];
  const float* img2 = (const float*)d_in[1];
  float* out     = (float*)d_out;
  float* partial = (float*)d_ws;        // NPART floats = 48 KB of scratch

  dim3 grid(TILES_X, TILES_Y, NIMG);    // 8 x 32 x 48 = 12288 blocks
  ssim_tile_kernel<<<grid, 256, 0, stream>>>(img1, img2, partial);
  ssim_finalize_kernel<<<1, 256, 0, stream>>>(partial, out);
}